// IDCT2_24816321037011
// MI455X (gfx1250) — compile-verified
//
#include <hip/hip_runtime.h>
#include <hip/hip_bf16.h>

// ---------------------------------------------------------------------------
// IDCT2 on 4096x4096 fp32  ==  out = C * x * C^T  with
//   C[n,k] = cos(pi * k * (2n+1) / 8192)
// Two fp32 WMMA GEMMs (v_wmma_f32_16x16x4_f32). The cosine matrix is stored
// k-major (Ct[k][n] = C[n][k]) so every tile load is coalesced. LDS tiles use
// an interleaved k-pair layout so each WMMA fragment is ONE aligned
// ds_load_b64 into an even VGPR pair (no repack movs).
// ---------------------------------------------------------------------------

typedef __attribute__((ext_vector_type(2))) float v2f;
typedef __attribute__((ext_vector_type(8))) float v8f;

#define L4096 4096
#define BM 128
#define BN 128
#define BK 16
// LDS: 8 k-pair rows x 128 cols x 2 (interleaved k parity) + pad.
// Row stride 288 words: 288 % 64 == 32, so the k2+1 half-wave's banks are
// shifted by 32 vs the k2 half-wave -> conflict-free fragment b64 loads.
#define LDP 288

// ---------------------------------------------------------------------------
// Ct[k][n] = cos(pi * k * (2n+1) / 8192), k-major, 4096x4096.
// Integer phase reduction: period of cos(pi*t/8192) is t=16384 and
// k*(2n+1) < 2^26 is exact in int32 -> angle < 2*pi, full fp32 accuracy.
// ---------------------------------------------------------------------------
__global__ __launch_bounds__(256) void build_ct_kernel(float* __restrict__ Ct) {
    const int idx = (blockIdx.x * 256 + threadIdx.x) * 4;   // 4 floats/thread
    const int k  = idx >> 12;
    const int n0 = idx & 4095;
    const float scale = 3.14159265358979323846f / 8192.0f;
    float4 v;
    #pragma unroll
    for (int j = 0; j < 4; ++j) {
        const int n = n0 + j;
        const int t = (k * (2 * n + 1)) & 16383;
        ((float*)&v)[j] = cosf((float)t * scale);
    }
    *(float4*)(Ct + idx) = v;
}

// ---------------------------------------------------------------------------
// Tiled WMMA GEMM, M=N=K=4096:
//   A_KXM == false :  D[m,n] = sum_k A[m*4096+k] * B[k*4096+n]   (A row-major)
//   A_KXM == true  :  D[m,n] = sum_k A[k*4096+m] * B[k*4096+n]   (A k-major)
// B is always k-major. 256 threads (8 wave32), 128x128 tile, BK=16.
// Wave grid 4(m) x 2(n): each wave owns 32x64 = 2x4 tiles of 16x16.
// ---------------------------------------------------------------------------
template <bool A_KXM>
__global__ __launch_bounds__(256) void gemm_wmma_f32_kernel(
    const float* __restrict__ A, const float* __restrict__ B,
    float* __restrict__ D)
{
    // Interleaved layout: element (k, col) at [ (k>>1)*LDP + col*2 + (k&1) ]
    __shared__ float As[8 * LDP];
    __shared__ float Bs[8 * LDP];

    const int tid  = threadIdx.x;
    const int lane = tid & 31;
    const int wid  = tid >> 5;            // 0..7
    const int lmod  = lane & 15;          // N / M index within 16x16 tile
    const int lhalf = lane >> 4;          // half-wave select
    const int wave_m = (wid & 3) * 32;    // 0,32,64,96
    const int wave_n = (wid >> 2) * 64;   // 0,64

    const int m0 = blockIdx.y * BM;
    const int n0 = blockIdx.x * BN;

    v8f acc[2][4];
    #pragma unroll
    for (int mt = 0; mt < 2; ++mt)
        #pragma unroll
        for (int nt = 0; nt < 4; ++nt)
            acc[mt][nt] = (v8f)0.0f;

    // Fill-path indices (k-major sources): one 2k x 4n chunk per thread.
    const int k2c  = tid >> 5;            // k-pair row 0..7
    const int ncol = (tid & 31) * 4;      // col start 0..124

    for (int kt = 0; kt < L4096; kt += BK) {
        // ---- A tile ----
        if (A_KXM) {
            // A[k][m] k-major: 2 x b128 global, 4 x b64 interleaved ds stores
            const float* src = A + (size_t)(kt + 2 * k2c) * L4096 + m0 + ncol;
            const float4 r0 = *(const float4*)src;
            const float4 r1 = *(const float4*)(src + L4096);
            #pragma unroll
            for (int j = 0; j < 4; ++j) {
                v2f p; p.x = ((const float*)&r0)[j]; p.y = ((const float*)&r1)[j];
                *(v2f*)&As[k2c * LDP + (ncol + j) * 2] = p;
            }
        } else {
            // A[m][k] (x): float4 along k -> two interleaved-pair b64 stores
            #pragma unroll
            for (int i = 0; i < 2; ++i) {
                const int idx = tid + i * 256;      // 0..511
                const int m   = idx >> 2;           // 0..127
                const int k4  = (idx & 3) * 4;      // 0,4,8,12
                const float4 v = *(const float4*)(A + (size_t)(m0 + m) * L4096 + kt + k4);
                v2f p0; p0.x = v.x; p0.y = v.y;
                v2f p1; p1.x = v.z; p1.y = v.w;
                *(v2f*)&As[((k4 >> 1) + 0) * LDP + m * 2] = p0;
                *(v2f*)&As[((k4 >> 1) + 1) * LDP + m * 2] = p1;
            }
        }
        // ---- B tile (always k-major) ----
        {
            const float* src = B + (size_t)(kt + 2 * k2c) * L4096 + n0 + ncol;
            const float4 r0 = *(const float4*)src;
            const float4 r1 = *(const float4*)(src + L4096);
            #pragma unroll
            for (int j = 0; j < 4; ++j) {
                v2f p; p.x = ((const float*)&r0)[j]; p.y = ((const float*)&r1)[j];
                *(v2f*)&Bs[k2c * LDP + (ncol + j) * 2] = p;
            }
        }
        __syncthreads();

        // ---- compute: 4 K-steps of 4, 8 wmma each ----
        #pragma unroll
        for (int kk = 0; kk < BK; kk += 4) {
            // fp32 frag layout (ISA 7.12.2): VGPR0 = K+0, VGPR1 = K+1,
            // half-wave (lhalf) selects K+2/K+3 -> k-pair row (kk>>1)+lhalf.
            const int krow2 = (kk >> 1) + lhalf;
            v2f afrag[2];
            #pragma unroll
            for (int mt = 0; mt < 2; ++mt) {
                const int m = wave_m + 16 * mt + lmod;
                afrag[mt] = *(const v2f*)&As[krow2 * LDP + m * 2];
            }
            v2f bfrag[4];
            #pragma unroll
            for (int nt = 0; nt < 4; ++nt) {
                const int n = wave_n + 16 * nt + lmod;
                bfrag[nt] = *(const v2f*)&Bs[krow2 * LDP + n * 2];
            }
            #pragma unroll
            for (int mt = 0; mt < 2; ++mt)
                #pragma unroll
                for (int nt = 0; nt < 4; ++nt)
                    acc[mt][nt] = __builtin_amdgcn_wmma_f32_16x16x4_f32(
                        /*neg_a=*/false, afrag[mt],
                        /*neg_b=*/false, bfrag[nt],
                        /*c_mod=*/(short)0, acc[mt][nt],
                        /*reuse_a=*/false, /*reuse_b=*/false);
        }
        __syncthreads();
    }

    // ---- epilogue: C/D layout -> lane (lmod) = N, VGPR v = M + 8*lhalf ----
    #pragma unroll
    for (int mt = 0; mt < 2; ++mt) {
        #pragma unroll
        for (int nt = 0; nt < 4; ++nt) {
            const int n = n0 + wave_n + 16 * nt + lmod;
            #pragma unroll
            for (int v = 0; v < 8; ++v) {
                const int m = m0 + wave_m + 16 * mt + v + 8 * lhalf;
                D[(size_t)m * L4096 + n] = acc[mt][nt][v];
            }
        }
    }
}

// ---------------------------------------------------------------------------
// Launch: build Ct (ws+0), T = x*Ct (ws+64MB), out = Ct^T * T.
// Needs 128 MB workspace. Stream-ordered; graph-capture safe.
// ---------------------------------------------------------------------------
extern "C" void kernel_launch(void* const* d_in, const int* in_sizes, int n_in,
                              void* d_out, int out_size, void* d_ws, size_t ws_size,
                              hipStream_t stream) {
    (void)in_sizes; (void)n_in; (void)out_size; (void)ws_size;
    const float* x = (const float*)d_in[0];          // 4096*4096 fp32
    float* Ct = (float*)d_ws;                                                   // 64 MB
    float* T  = (float*)((char*)d_ws + (size_t)L4096 * L4096 * sizeof(float));  // 64 MB

    // 1) cosine matrix, k-major
    build_ct_kernel<<<(L4096 * L4096 / 4) / 256, 256, 0, stream>>>(Ct);

    dim3 grid(L4096 / BN, L4096 / BM);
    // 2) T = x * Ct        (A row-major)
    gemm_wmma_f32_kernel<false><<<grid, 256, 0, stream>>>(x, Ct, T);
    // 3) out = Ct^T * T    (A k-major)
    gemm_wmma_f32_kernel<true><<<grid, 256, 0, stream>>>(Ct, T, (float*)d_out);
}